// MemoryMoudle_69853348102294
// MI455X (gfx1250) — compile-verified
//
#include <hip/hip_runtime.h>
#include <math.h>

#define N_MEM 10

// ---- problem geometry (floats) ----
#define D1 (8*64*64*64)     // 2,097,152 per item, feature1
#define D2 (8*128*32*32)    // 1,048,576
#define D3 (8*256*16*16)    //   524,288
#define N4_1 (D1/4)
#define N4_2 (D2/4)
#define N4_3 (D3/4)
#define HALF4_1 (64*64*64/4)   // per-batch half (channels of one source) in float4
#define HALF4_2 (128*32*32/4)
#define HALF4_3 (256*16*16/4)
#define TOT4_CI1 (8*2*HALF4_1) // 1,048,576 float4
#define TOT4_CI2 (8*2*HALF4_2) //   524,288
#define TOT4_CI3 (8*2*HALF4_3) //   262,144
// output float offsets (return order: CI1,CI2,CI3,mi1,mi2,mi3)
#define OFF_CI1 0
#define OFF_CI2 (2*D1)
#define OFF_CI3 (2*D1 + 2*D2)
#define OFF_MI1 (2*D1 + 2*D2 + 2*D3)
#define OFF_MI2 (OFF_MI1 + D1)
#define OFF_MI3 (OFF_MI2 + D2)

typedef __attribute__((ext_vector_type(2))) float v2f;
typedef __attribute__((ext_vector_type(8))) float v8f;

// Layout-robust wave32 sum using V_WMMA_F32_16X16X4_F32.
// A = all-ones matrix (all-ones under ANY operand layout).
// B holds each lane's partial p twice (64 lane-slots <-> 64 matrix cells, bijective),
// so D[m][n] = colsum(B)[n] for every row m, hence sum over any full rows of D
// equals rows * (2 * sum(p)).  c[0] across all 32 lanes covers rows 0 and 8 of the
// documented C layout -> Sum_lanes(c[0]) = 4 * sum(p).
__device__ __forceinline__ float wave_sum_wmma(float p) {
  v2f a; a[0] = 1.0f; a[1] = 1.0f;
  v2f b; b[0] = p;    b[1] = p;
  v8f c = {};
  c = __builtin_amdgcn_wmma_f32_16x16x4_f32(false, a, false, b, (short)0, c, false, false);
  float r = c[0];
  r += __shfl_xor(r, 16, 32);
  r += __shfl_xor(r,  8, 32);
  r += __shfl_xor(r,  4, 32);
  r += __shfl_xor(r,  2, 32);
  r += __shfl_xor(r,  1, 32);
  return r * 0.25f;
}

// Zero the 3x16 double accumulators (workspace is poisoned by the harness).
__global__ void init_ws_kernel(double* __restrict__ acc) {
  int t = threadIdx.x;
  if (t < 48) acc[t] = 0.0;
}

// One scale: stream feature once per thread, loop all 10 memory items at the same
// offset.  blockDim must be 256 (8 waves).  acc = 10 (padded 16) f64 accumulators.
__global__ void dist_kernel(const float* __restrict__ f, const float* __restrict__ M,
                            int n4, double* __restrict__ acc) {
  __shared__ float partial[8][N_MEM];
  float s[N_MEM];
#pragma unroll
  for (int i = 0; i < N_MEM; ++i) s[i] = 0.0f;

  const float4* __restrict__ f4 = (const float4*)f;
  const float4* __restrict__ m4 = (const float4*)M;
  const int stride = gridDim.x * blockDim.x;
  for (int j = blockIdx.x * blockDim.x + threadIdx.x; j < n4; j += stride) {
    float4 fv = f4[j];
#pragma unroll
    for (int i = 0; i < N_MEM; ++i) {
      float4 mv = m4[(size_t)i * (size_t)n4 + (size_t)j];
      float dx = fv.x - mv.x, dy = fv.y - mv.y, dz = fv.z - mv.z, dw = fv.w - mv.w;
      s[i] = fmaf(dx, dx, s[i]);
      s[i] = fmaf(dy, dy, s[i]);
      s[i] = fmaf(dz, dz, s[i]);
      s[i] = fmaf(dw, dw, s[i]);
    }
  }

  const int lane = threadIdx.x & 31;
  const int wave = threadIdx.x >> 5;
#pragma unroll
  for (int i = 0; i < N_MEM; ++i) {
    float r = wave_sum_wmma(s[i]);          // EXEC is full here (uniform post-loop)
    if (lane == 0) partial[wave][i] = r;
  }
  __syncthreads();
  if (threadIdx.x < N_MEM) {
    float sum = 0.0f;
#pragma unroll
    for (int w = 0; w < 8; ++w) sum += partial[w][threadIdx.x];
    atomicAdd(&acc[threadIdx.x], (double)sum);   // global_atomic_add_f64
  }
}

// total[i] = sqrt(s1)+sqrt(s2)+sqrt(s3); first argmin -> *idxOut
__global__ void argmin_kernel(const double* __restrict__ acc, int* __restrict__ idxOut) {
  if (threadIdx.x == 0 && blockIdx.x == 0) {
    double best = 1.0e300;
    int bi = 0;
    for (int i = 0; i < N_MEM; ++i) {
      double t = sqrt(acc[i]) + sqrt(acc[16 + i]) + sqrt(acc[32 + i]);
      if (t < best) { best = t; bi = i; }
    }
    *idxOut = bi;
  }
}

// CI = concat([feature, M[idx]], axis=1): per batch b, first half4 float4s from
// feature, next half4 from the selected memory slice.
__global__ void concat_kernel(float4* __restrict__ dst, const float4* __restrict__ f,
                              const float4* __restrict__ M, const int* __restrict__ idxPtr,
                              int half4, int total4, int d4) {
  const int idx = *idxPtr;
  const float4* __restrict__ mi = M + (size_t)idx * (size_t)d4;
  const int stride = gridDim.x * blockDim.x;
  const int per_b = 2 * half4;
  for (int t = blockIdx.x * blockDim.x + threadIdx.x; t < total4; t += stride) {
    int b = t / per_b;
    int r = t - b * per_b;
    float4 v;
    if (r < half4) v = f[(size_t)b * half4 + r];
    else           v = mi[(size_t)b * half4 + (r - half4)];
    dst[t] = v;
  }
}

__global__ void copy_mi_kernel(float4* __restrict__ dst, const float4* __restrict__ M,
                               const int* __restrict__ idxPtr, int d4) {
  const int idx = *idxPtr;
  const float4* __restrict__ mi = M + (size_t)idx * (size_t)d4;
  const int stride = gridDim.x * blockDim.x;
  for (int t = blockIdx.x * blockDim.x + threadIdx.x; t < d4; t += stride)
    dst[t] = mi[t];
}

extern "C" void kernel_launch(void* const* d_in, const int* in_sizes, int n_in,
                              void* d_out, int out_size, void* d_ws, size_t ws_size,
                              hipStream_t stream) {
  const float* f1 = (const float*)d_in[0];
  const float* f2 = (const float*)d_in[1];
  const float* f3 = (const float*)d_in[2];
  const float* M1 = (const float*)d_in[3];
  const float* M2 = (const float*)d_in[4];
  const float* M3 = (const float*)d_in[5];
  float* out = (float*)d_out;

  double* acc = (double*)d_ws;                       // 3 x 16 doubles (384 B)
  int* idxp   = (int*)((char*)d_ws + 48 * sizeof(double));

  (void)in_sizes; (void)n_in; (void)out_size; (void)ws_size;

  // 1) zero accumulators
  hipLaunchKernelGGL(init_ws_kernel, dim3(1), dim3(64), 0, stream, acc);

  // 2) distances per scale (256 blocks x 256 threads divides all n4 exactly)
  hipLaunchKernelGGL(dist_kernel, dim3(256), dim3(256), 0, stream, f1, M1, N4_1, acc + 0);
  hipLaunchKernelGGL(dist_kernel, dim3(256), dim3(256), 0, stream, f2, M2, N4_2, acc + 16);
  hipLaunchKernelGGL(dist_kernel, dim3(256), dim3(256), 0, stream, f3, M3, N4_3, acc + 32);

  // 3) argmin
  hipLaunchKernelGGL(argmin_kernel, dim3(1), dim3(32), 0, stream, acc, idxp);

  // 4) outputs: CI1, CI2, CI3, mi1, mi2, mi3
  hipLaunchKernelGGL(concat_kernel, dim3(2048), dim3(256), 0, stream,
                     (float4*)(out + OFF_CI1), (const float4*)f1, (const float4*)M1,
                     idxp, HALF4_1, TOT4_CI1, N4_1);
  hipLaunchKernelGGL(concat_kernel, dim3(1024), dim3(256), 0, stream,
                     (float4*)(out + OFF_CI2), (const float4*)f2, (const float4*)M2,
                     idxp, HALF4_2, TOT4_CI2, N4_2);
  hipLaunchKernelGGL(concat_kernel, dim3(512), dim3(256), 0, stream,
                     (float4*)(out + OFF_CI3), (const float4*)f3, (const float4*)M3,
                     idxp, HALF4_3, TOT4_CI3, N4_3);
  hipLaunchKernelGGL(copy_mi_kernel, dim3(1024), dim3(256), 0, stream,
                     (float4*)(out + OFF_MI1), (const float4*)M1, idxp, N4_1);
  hipLaunchKernelGGL(copy_mi_kernel, dim3(512), dim3(256), 0, stream,
                     (float4*)(out + OFF_MI2), (const float4*)M2, idxp, N4_2);
  hipLaunchKernelGGL(copy_mi_kernel, dim3(256), dim3(256), 0, stream,
                     (float4*)(out + OFF_MI3), (const float4*)M3, idxp, N4_3);
}